// CutLayer_27728308863382
// MI455X (gfx1250) — compile-verified
//
#include <hip/hip_runtime.h>
#include <math.h>

#define E 51
#define NBINS 50

typedef float v2f __attribute__((ext_vector_type(2)));
typedef float v8f __attribute__((ext_vector_type(8)));

#ifndef __has_builtin
#define __has_builtin(x) 0
#endif
#if __has_builtin(__builtin_amdgcn_wmma_f32_16x16x4_f32)
#define USE_WMMA 1
#else
#define USE_WMMA 0
#endif

// ---------------------------------------------------------------------------
// Kernel 1: per-block min/max of feature 0 (rows are float4; .x is feature 0)
// ---------------------------------------------------------------------------
__global__ void k_minmax(const float4* __restrict__ in, int n,
                         float* __restrict__ bmin, float* __restrict__ bmax) {
    __shared__ float smin[256], smax[256];
    int tid = threadIdx.x;
    int stride = gridDim.x * blockDim.x;
    float vmin = __builtin_inff(), vmax = -__builtin_inff();
    for (int i = blockIdx.x * blockDim.x + tid; i < n; i += stride) {
        float x = in[i].x;
        vmin = fminf(vmin, x);
        vmax = fmaxf(vmax, x);
    }
    smin[tid] = vmin; smax[tid] = vmax;
    __syncthreads();
    for (int s = 128; s > 0; s >>= 1) {
        if (tid < s) {
            smin[tid] = fminf(smin[tid], smin[tid + s]);
            smax[tid] = fmaxf(smax[tid], smax[tid + s]);
        }
        __syncthreads();
    }
    if (tid == 0) { bmin[blockIdx.x] = smin[0]; bmax[blockIdx.x] = smax[0]; }
}

// ---------------------------------------------------------------------------
// Kernel 2: reduce partial min/max, build edges, zero global histograms
// ---------------------------------------------------------------------------
__global__ void k_edges(const float* __restrict__ bmin, const float* __restrict__ bmax,
                        int nb, float* __restrict__ gmm, float* __restrict__ edges,
                        unsigned long long* __restrict__ zero64, int nz) {
    __shared__ float smin[256], smax[256];
    int tid = threadIdx.x;
    float vmin = __builtin_inff(), vmax = -__builtin_inff();
    for (int i = tid; i < nb; i += 256) {
        vmin = fminf(vmin, bmin[i]);
        vmax = fmaxf(vmax, bmax[i]);
    }
    smin[tid] = vmin; smax[tid] = vmax;
    __syncthreads();
    for (int s = 128; s > 0; s >>= 1) {
        if (tid < s) {
            smin[tid] = fminf(smin[tid], smin[tid + s]);
            smax[tid] = fmaxf(smax[tid], smax[tid + s]);
        }
        __syncthreads();
    }
    float mn = smin[0], mx = smax[0];
    if (tid == 0) { gmm[0] = mn; gmm[1] = mx; }
    float step = (mx - mn) / (float)NBINS;
    if (tid < E) edges[tid] = (tid == NBINS) ? mx : mn + (float)tid * step;
    for (int i = tid; i < nz; i += 256) zero64[i] = 0ull;   // re-zeroed every replay
}

// ---------------------------------------------------------------------------
// Kernel 3: histogram of j_r = #{edges < x}, class-packed in u64
// (low32 = all samples, high32 = y==1; counts <= 8M so halves never carry).
// The "<=" vs "<" distinction is recovered from a rare exact-tie counter:
//   l[e]    = r[e]    - #{x == edges[e]}
//   ns_lt[e]= ns_le[e]- #{y==1, x == edges[e]}
// Per-wave LDS histogram copies (wave32 -> tid>>5) kill inter-wave contention.
// ---------------------------------------------------------------------------
__global__ void k_hist(const float4* __restrict__ in, const int* __restrict__ tgt, int n,
                       const float* __restrict__ edges,
                       unsigned long long* __restrict__ hr_g,
                       unsigned long long* __restrict__ tie_g) {
    __shared__ float se[64];                       // edges padded with +inf
    __shared__ unsigned long long hc[8 * 52];      // 8 wave-private copies
    __shared__ unsigned long long th[52];          // shared tie counters (rare)
    int tid = threadIdx.x;
    int wid = tid >> 5;                            // wave32
    for (int i = tid; i < 64; i += blockDim.x)
        se[i] = (i < E) ? edges[i] : __builtin_inff();
    for (int i = tid; i < 8 * 52; i += blockDim.x) hc[i] = 0ull;
    for (int i = tid; i < 52; i += blockDim.x) th[i] = 0ull;
    __syncthreads();

    unsigned long long* myh = &hc[wid * 52];
    int stride = gridDim.x * blockDim.x;
    for (int i = blockIdx.x * blockDim.x + tid; i < n; i += stride) {
        __builtin_prefetch(&in[i + stride], 0, 1);   // -> global_prefetch_b8
        float x = in[i].x;
        int y = tgt[i];
        // branchless lower_bound over 64 sorted entries: c = #{se < x} (<= 50)
        int c = 0;
        if (se[31]     < x) c  = 32;
        if (se[c + 15] < x) c += 16;
        if (se[c + 7]  < x) c += 8;
        if (se[c + 3]  < x) c += 4;
        if (se[c + 1]  < x) c += 2;
        if (se[c]      < x) c += 1;
        unsigned long long inc = 1ull + ((unsigned long long)(y == 1) << 32);
        atomicAdd(&myh[c], inc);                    // ds_add_u64, 1 per element
        if (se[c] == x) atomicAdd(&th[c], inc);     // exact tie: rare
    }
    __syncthreads();
    for (int b = tid; b < 52; b += blockDim.x) {
        unsigned long long s = 0ull;
        #pragma unroll
        for (int w = 0; w < 8; ++w) s += hc[w * 52 + b];
        if (s) atomicAdd(&hr_g[b], s);
        unsigned long long t = th[b];
        if (t) atomicAdd(&tie_g[b], t);
    }
}

// ---------------------------------------------------------------------------
// Kernel 4: single wave32. Cumsums, candidate mask, then the 51x51 pair loss.
// c2(i,j), c3(i,j) are outer sums -> built with v_wmma_f32_16x16x4_f32.
// All counts are integers < 2^24, so every f32 sum is exact (matches ref).
// ---------------------------------------------------------------------------
__global__ void k_pairs(const unsigned long long* __restrict__ hr_g,
                        const unsigned long long* __restrict__ tie_g,
                        const float* __restrict__ edges, int n,
                        float* __restrict__ sel, unsigned* __restrict__ casep) {
    __shared__ float t0s[64], t1s[64], rc2[64], cc2[64], rc3[64], cc3[64], eds[64];
    __shared__ int   msk[64];
    __shared__ float redV[32];
    __shared__ int   redF[32], redC[32];
    __shared__ float sNF, sA, sB;

    int lane = threadIdx.x;
    if (lane == 0) {
        float nf = (float)n;
        float aL = -log1pf(-1e-7f);
        float bL = -logf(1e-7f);
        sNF = nf; sA = aL; sB = bL;
        float ns_le[E], ns_lt[E], nb_le[E], nb_lt[E];
        float cr = 0.f, crs = 0.f;
        for (int e = 0; e < E; ++e) {
            unsigned long long hv = hr_g[e];
            unsigned long long tv = tie_g[e];
            cr  += (float)(unsigned)(hv & 0xffffffffull);
            crs += (float)(unsigned)(hv >> 32);
            float tA = (float)(unsigned)(tv & 0xffffffffull);
            float tS = (float)(unsigned)(tv >> 32);
            ns_le[e] = crs;            nb_le[e] = cr - crs;
            float l  = cr - tA;        float nslt = crs - tS;
            ns_lt[e] = nslt;           nb_lt[e] = l - nslt;
        }
        float Ns = ns_le[E - 1];
        float Nb = nf - Ns;
        float h0[NBINS], h1[NBINS];
        for (int k = 0; k < NBINS; ++k) {
            h0[k] = nb_le[k + 1] - nb_lt[k];
            h1[k] = ns_le[k + 1] - ns_lt[k];
        }
        for (int e = 0; e < 64; ++e) msk[e] = 0;
        int cnt = 0;
        for (int k = 0; k < NBINS - 1; ++k) {
            bool g0a = h0[k] > h1[k],     g0b = h0[k + 1] > h1[k + 1];
            bool g1a = h1[k] > h0[k],     g1b = h1[k + 1] > h0[k + 1];
            bool cnd0 = (g0a != g0b) && (h0[k] > 0.f);
            bool cnd1 = (g1a != g1b) && (h1[k] > 0.f);
            if (cnd0 || cnd1) { msk[k + 1] = 1; ++cnt; }
        }
        if (cnt == 1) msk[E - 1] = 1;
        for (int e = 0; e < 64; ++e) {
            if (e < E) {
                float c0v = ns_le[e] + (Nb - nb_le[e]);
                float c1v = Ns - ns_lt[e] + nb_lt[e];
                t0s[e] = ((nf - c0v) * bL + c0v * aL) / nf;
                t1s[e] = ((nf - c1v) * bL + c1v * aL) / nf;
                rc2[e] = (nb_lt[e] - ns_lt[e]) + Nb;   // i-part of c2 (+Nb)
                cc2[e] = ns_le[e] - nb_le[e];          // j-part of c2
                rc3[e] = ns_le[e] - nb_lt[e];          // i-part of c3
                cc3[e] = nb_le[e] - ns_lt[e] + Ns;     // j-part of c3 (+Ns)
                eds[e] = edges[e];
            } else {
                t0s[e] = 0.f; t1s[e] = 0.f;
                rc2[e] = 0.f; cc2[e] = 0.f; rc3[e] = 0.f; cc3[e] = 0.f;
                eds[e] = 0.f;
            }
        }
    }
    __syncthreads();

    float nf = sNF, aL = sA, bL = sB;
    float bestV = __builtin_inff();
    int   bestF = 0x7fffffff, bestC = 0;
    int half = lane >> 4, lm = lane & 15;

    for (int ti = 0; ti < 4; ++ti) {
        for (int tj = 0; tj < 4; ++tj) {
            float d2a[8], d3a[8];
#if USE_WMMA
            // A (16x4): col0 = row-part, col1 = 1, cols 2..3 = 0
            // B (4x16): row0 = 1, row1 = col-part, rows 2..3 = 0
            v2f a2, b2, a3, b3;
            a2.x = half ? 0.f : rc2[ti * 16 + lm];  a2.y = half ? 0.f : 1.f;
            b2.x = half ? 0.f : 1.f;                b2.y = half ? 0.f : cc2[tj * 16 + lm];
            a3.x = half ? 0.f : rc3[ti * 16 + lm];  a3.y = half ? 0.f : 1.f;
            b3.x = half ? 0.f : 1.f;                b3.y = half ? 0.f : cc3[tj * 16 + lm];
            v8f z = {0.f, 0.f, 0.f, 0.f, 0.f, 0.f, 0.f, 0.f};
            v8f d2 = __builtin_amdgcn_wmma_f32_16x16x4_f32(false, a2, false, b2,
                                                           (short)0, z, false, false);
            v8f d3 = __builtin_amdgcn_wmma_f32_16x16x4_f32(false, a3, false, b3,
                                                           (short)0, z, false, false);
            #pragma unroll
            for (int r = 0; r < 8; ++r) { d2a[r] = d2[r]; d3a[r] = d3[r]; }
#else
            #pragma unroll
            for (int r = 0; r < 8; ++r) {
                int gi = ti * 16 + r + 8 * half, gj = tj * 16 + lm;
                d2a[r] = rc2[gi] + cc2[gj];
                d3a[r] = rc3[gi] + cc3[gj];
            }
#endif
            #pragma unroll
            for (int r = 0; r < 8; ++r) {
                int gi = ti * 16 + r + 8 * half;   // C/D layout: M = r + 8*(lane>=16)
                int gj = tj * 16 + lm;             // N = lane % 16
                float c2v = d2a[r], c3v = d3a[r];
                float L2 = ((nf - c2v) * bL + c2v * aL) / nf;
                float L3 = ((nf - c3v) * bL + c3v * aL) / nf;
                float L0 = t0s[gi], L1 = t1s[gi];
                float pb = L0; int pc = 0;
                if (L1 < pb) { pb = L1; pc = 1; }
                if (L2 < pb) { pb = L2; pc = 2; }
                if (L3 < pb) { pb = L3; pc = 3; }
                bool inb = (gi < E) && (gj < E);
                bool valid = inb && (gi < gj) && msk[gi] && msk[gj];
                float v = valid ? pb : __builtin_inff();
                int flat = inb ? (gi * E + gj) : 0x7fffffff;
                if ((v < bestV) || (v == bestV && flat < bestF)) {
                    bestV = v; bestF = flat; bestC = pc;
                }
            }
        }
    }

    redV[lane] = bestV; redF[lane] = bestF; redC[lane] = bestC;
    __syncthreads();
    if (lane == 0) {
        float bv = redV[0]; int bf = redF[0], bc = redC[0];
        for (int l = 1; l < 32; ++l) {
            if ((redV[l] < bv) || (redV[l] == bv && redF[l] < bf)) {
                bv = redV[l]; bf = redF[l]; bc = redC[l];
            }
        }
        if (bf == 0x7fffffff) bf = 0;   // defensive; (0,0) entry always exists
        int i = bf / E, j = bf % E;
        sel[0] = eds[i];
        sel[1] = eds[j];
        casep[0] = (unsigned)bc;
    }
}

// ---------------------------------------------------------------------------
// Kernel 5: streaming prediction
// ---------------------------------------------------------------------------
__global__ void k_pred(const float4* __restrict__ in, int n, const float* __restrict__ sel,
                       const unsigned* __restrict__ casep, int* __restrict__ out) {
    float lower = sel[0], upper = sel[1];
    unsigned c = casep[0];
    int stride = gridDim.x * blockDim.x;
    for (int i = blockIdx.x * blockDim.x + threadIdx.x; i < n; i += stride) {
        __builtin_prefetch(&in[i + stride], 0, 1);
        float x = in[i].x;
        bool p0 = (x <= lower);
        bool p1 = (x >= lower);
        bool p;
        if (c == 0)      p = p0;
        else if (c == 1) p = p1;
        else if (c == 2) p = p1 && (x <= upper);
        else             p = p0 || (x >= upper);
        out[i] = p ? 1 : 0;
    }
}

// ---------------------------------------------------------------------------
extern "C" void kernel_launch(void* const* d_in, const int* in_sizes, int n_in,
                              void* d_out, int out_size, void* d_ws, size_t ws_size,
                              hipStream_t stream) {
    const float* inp = (const float*)d_in[0];     // [N,4] float32
    const int*   tgt = (const int*)d_in[1];       // [N]   int32
    int n = in_sizes[1];                          // N = 8,000,000
    const float4* in4 = (const float4*)inp;
    int* out = (int*)d_out;

    // workspace layout
    float*              wf    = (float*)d_ws;
    unsigned*           wu    = (unsigned*)d_ws;
    unsigned long long* w8    = (unsigned long long*)d_ws;
    float*    bmin  = wf;               // [0..511]
    float*    bmax  = wf + 512;         // [512..1023]
    float*    gmm   = wf + 1024;        // [1024..1025]
    float*    edges = wf + 1026;        // [1026..1076]
    float*    sel   = wf + 1078;        // lower, upper
    unsigned* casep = wu + 1080;        // case id
    unsigned long long* hr_g  = w8 + 544;   // bytes 4352.. : 52 u64
    unsigned long long* tie_g = w8 + 596;   // 52 u64

    k_minmax<<<512, 256, 0, stream>>>(in4, n, bmin, bmax);
    k_edges <<<1,   256, 0, stream>>>(bmin, bmax, 512, gmm, edges, hr_g, 104);
    k_hist  <<<1024,256, 0, stream>>>(in4, tgt, n, edges, hr_g, tie_g);
    k_pairs <<<1,   32,  0, stream>>>(hr_g, tie_g, edges, n, sel, casep);
    k_pred  <<<2048,256, 0, stream>>>(in4, n, sel, casep, out);
}